// RNN_13907104105208
// MI455X (gfx1250) — compile-verified
//
#include <hip/hip_runtime.h>

// Problem constants (from the reference)
constexpr int KE = 8192;   // input feature dim
constexpr int KH = 128;    // hidden dim
constexpr int KO = 2;      // output dim
constexpr int KT = 4096;   // time steps
constexpr int KL = 64;     // chunk length
constexpr int KC = 64;     // number of chunks (KT / KL)

typedef __attribute__((ext_vector_type(2))) float v2f;
typedef __attribute__((ext_vector_type(8))) float v8f;

// CDNA5 native fp32 tensor op: D(16x16) = A(16x4) * B(4x16) + C
__device__ __forceinline__ v8f wmma_f32_16x16x4(v2f a, v2f b, v8f c) {
  return __builtin_amdgcn_wmma_f32_16x16x4_f32(
      false, a, false, b, (short)0, c, false, false);
}

// ---------------------------------------------------------------------------
// Tensor Data Mover: DMA one 16-row x 64-col fp32 tile (row stride KE elems)
// from global memory into LDS, with HW padding of 4 DWORDs after every 64
// DWORDs so the LDS tile lands with a 68-float row stride (conflict-free for
// the WMMA A-operand read pattern). Issued by one wave; tracked on TENSORcnt.
// D# layout per CDNA5 ISA ch.8 (async_tensor).
// ---------------------------------------------------------------------------
__device__ __forceinline__ void tdm_load_tile16x64(unsigned lds_addr,
                                                   const float* gptr) {
  typedef unsigned int v4u __attribute__((ext_vector_type(4)));
  typedef int v8i __attribute__((ext_vector_type(8)));
  typedef int v4i __attribute__((ext_vector_type(4)));
  const unsigned long long ga = (unsigned long long)(uintptr_t)gptr;

  v4u g0;
  g0[0] = 1u;                                   // count=1 (valid), user mode
  g0[1] = lds_addr;                             // lds_addr (bytes) [63:32]
  g0[2] = (unsigned)(ga & 0xFFFFFFFFu);         // global_addr [95:64]
  g0[3] = (unsigned)((ga >> 32) & 0x1FFFFFFu)   // global_addr [120:96]
        | (2u << 30);                           // type=2 ("image") [127:126]

  v8i g1;
  g1[0] = (2 << 16)        // data_size = 2 -> 4 bytes
        | (1 << 20)        // pad_enable
        | (5 << 22)        // pad_interval = 5 -> every 64 DWORDs
        | (3 << 25);       // pad_amount  = 3 -> 4 DWORDs => 68-float stride
  g1[1] = (KE & 0xFFFF) << 16;   // tensor_dim0 = 8192 (low16); ab_addr = 0
  g1[2] = 0;                     // tensor_dim0 hi16 = 0; tensor_dim1 lo16 = 0
  g1[3] = (64 << 16) | 0x10;     // tile_dim0 = 64 | tensor_dim1 = 1<<20 (hi16)
  g1[4] = 16;                    // tile_dim1 = 16 rows; tile_dim2 = 0 (unused)
  g1[5] = KE;                    // tensor_dim0_stride = 8192 elems (lo32)
  g1[6] = 0;                     // stride hi16 = 0; tensor_dim1_stride = 0
  g1[7] = 0;

  v4i z4 = {0, 0, 0, 0};         // groups 2/3: unused (2D tile)
#if defined(__clang_major__) && __clang_major__ >= 23
  v8i z8 = {0, 0, 0, 0, 0, 0, 0, 0};
  __builtin_amdgcn_tensor_load_to_lds(g0, g1, z4, z4, z8, 0);
#else
  __builtin_amdgcn_tensor_load_to_lds(g0, g1, z4, z4, 0);
#endif
}

// ---------------------------------------------------------------------------
// K1: x_proj[T x H] = document[T x E] @ Wx[E x H] + b1
// 256 blocks, each owns a 16-row slab of T and all 128 columns (8 waves x 16).
// Memory-bound (128 MB doc traffic, ~5.5us floor at 23.3 TB/s). Document
// tiles are DMA'd into double-buffered LDS by the TDM, overlapping WMMA.
// ---------------------------------------------------------------------------
__global__ void __launch_bounds__(256) xproj_kernel(
    const float* __restrict__ doc, const float* __restrict__ W1,
    const float* __restrict__ b1, float* __restrict__ xproj) {
  __shared__ float tileA[2][16][68];  // stride 68 via TDM pad -> conflict-free
  const int tid  = threadIdx.x;
  const int wave = tid >> 5;
  const int lane = tid & 31;
  const int nl   = lane & 15;   // N within tile / A row (M)
  const int hi   = lane >> 4;   // half-wave select
  const int koff = hi * 2;      // A/B K sub-offset for upper half
  const int n0   = wave * 16;   // this wave's output column tile
  const int tile = blockIdx.x;  // 16-row slab of T

  const float* dbase = doc + (size_t)tile * 16 * KE;
  const unsigned ldsA0 = (unsigned)(uintptr_t)&tileA[0][0][0];
  const unsigned ldsA1 = (unsigned)(uintptr_t)&tileA[1][0][0];

  if (wave == 0) tdm_load_tile16x64(ldsA0, dbase);  // k-tile 0 in flight

  v8f acc = {0.f, 0.f, 0.f, 0.f, 0.f, 0.f, 0.f, 0.f};
  const int NT = KE / 64;
  for (int kt = 0; kt < NT; ++kt) {
    const int cur = kt & 1;
    if (wave == 0) __builtin_amdgcn_s_wait_tensorcnt((short)0);  // tile kt done
    __syncthreads();  // publish tile kt; all waves past tile kt-1's buffer
    if (wave == 0 && kt + 1 < NT)
      tdm_load_tile16x64(cur ? ldsA0 : ldsA1, dbase + (size_t)(kt + 1) * 64);
    if (kt + 1 < NT)  // warm next Wx panel in L2 (global_prefetch_b8)
      __builtin_prefetch(W1 + (size_t)(kt + 1) * 64 * KH + tid * 32, 0, 1);

    const int k0 = kt * 64;
#pragma unroll
    for (int kk = 0; kk < 64; kk += 4) {
      v2f a;
      a.x = tileA[cur][nl][kk + koff];
      a.y = tileA[cur][nl][kk + koff + 1];
      const int kg = k0 + kk + koff;
      v2f b;
      b.x = W1[(size_t)kg * KH + n0 + nl];        // coalesced over lanes
      b.y = W1[(size_t)(kg + 1) * KH + n0 + nl];
      acc = wmma_f32_16x16x4(a, b, acc);
    }
  }

  const float bn = b1[n0 + nl];
#pragma unroll
  for (int v = 0; v < 8; ++v) {  // C layout: VGPR v -> M = v + 8*hi, lane -> N
    const int m = v + hi * 8;
    xproj[((size_t)tile * 16 + m) * KH + n0 + nl] = acc[v] + bn;
  }
}

// ---------------------------------------------------------------------------
// K2a: chunk-parallel scan. 4 blocks x 16 chunks each.
// Block state P[16 x 128] in LDS; per step j: P = P @ Wh + X[j-rows],
// with the WMMA C accumulator initialized directly from x_proj.
// 64 sequential steps instead of 4096.
// ---------------------------------------------------------------------------
__global__ void __launch_bounds__(256) chunk_scan_kernel(
    const float* __restrict__ xproj, const float* __restrict__ W1,
    float* __restrict__ Pws) {
  __shared__ float P[16][132];  // stride 132 -> conflict-free A reads
  const int tid  = threadIdx.x;
  const int wave = tid >> 5;
  const int lane = tid & 31;
  const int nl   = lane & 15;
  const int hi   = lane >> 4;
  const int koff = hi * 2;
  const int n0   = wave * 16;
  const int blk  = blockIdx.x;  // 0..3, owns chunks [blk*16, blk*16+16)
  const float* Wh = W1 + (size_t)KE * KH;  // W1[E:] rows

  for (int i = tid; i < 16 * 132; i += 256) (&P[0][0])[i] = 0.f;
  __syncthreads();

  for (int j = 0; j < KL; ++j) {
    v8f c = {0.f, 0.f, 0.f, 0.f, 0.f, 0.f, 0.f, 0.f};
#pragma unroll
    for (int v = 0; v < 8; ++v) {  // C init = X[j] rows for this block's chunks
      const int m = v + hi * 8;
      const int t = (blk * 16 + m) * KL + j;
      c[v] = xproj[(size_t)t * KH + n0 + nl];
    }
#pragma unroll
    for (int k = 0; k < KH; k += 4) {
      v2f a;
      a.x = P[nl][k + koff];
      a.y = P[nl][k + koff + 1];
      v2f b;
      b.x = Wh[(size_t)(k + koff) * KH + n0 + nl];
      b.y = Wh[(size_t)(k + koff + 1) * KH + n0 + nl];
      c = wmma_f32_16x16x4(a, b, c);
    }
    __syncthreads();  // all waves done reading old P
#pragma unroll
    for (int v = 0; v < 8; ++v) P[v + hi * 8][n0 + nl] = c[v];
    __syncthreads();  // new P visible
  }
  for (int i = tid; i < 16 * KH; i += 256) {
    const int m = i >> 7, n = i & 127;
    Pws[((size_t)(blk * 16 + m)) * KH + n] = P[m][n];
  }
}

// ---------------------------------------------------------------------------
// K2b: single block. (a) Wh^64 via 6 WMMA squarings in ping-ponged dynamic LDS
// (135 KB, legal on 320 KB/WGP CDNA5); (b) h = h@Wh^64 + p_c over 64 chunks;
// (c) out = h @ W2 + b2.
// ---------------------------------------------------------------------------
__global__ void __launch_bounds__(256) combine_kernel(
    const float* __restrict__ W1, const float* __restrict__ Pws,
    const float* __restrict__ W2, const float* __restrict__ b2,
    float* __restrict__ out) {
  extern __shared__ float lds[];
  float* bufA = lds;                   // 128*132
  float* bufB = lds + 128 * 132;       // 128*132
  float* hv   = lds + 2 * 128 * 132;   // 128
  const int tid  = threadIdx.x;
  const int wave = tid >> 5;
  const int lane = tid & 31;
  const int nl   = lane & 15;
  const int hi   = lane >> 4;
  const int koff = hi * 2;
  const int n0   = wave * 16;
  const float* Wh = W1 + (size_t)KE * KH;

  for (int i = tid; i < KH * KH; i += 256) {
    const int r = i >> 7, c2 = i & 127;
    bufA[r * 132 + c2] = Wh[i];
  }
  __syncthreads();

  float* src = bufA;  // holds Wh^(2^s)
  float* dst = bufB;
  for (int s = 0; s < 6; ++s) {
    for (int mt = 0; mt < 8; ++mt) {  // each wave: column tile n0, all row tiles
      v8f c = {0.f, 0.f, 0.f, 0.f, 0.f, 0.f, 0.f, 0.f};
#pragma unroll
      for (int k = 0; k < KH; k += 4) {
        const int am = mt * 16 + nl;
        v2f a;
        a.x = src[am * 132 + k + koff];
        a.y = src[am * 132 + k + koff + 1];
        v2f b;
        b.x = src[(k + koff) * 132 + n0 + nl];
        b.y = src[(k + koff + 1) * 132 + n0 + nl];
        c = wmma_f32_16x16x4(a, b, c);
      }
#pragma unroll
      for (int v = 0; v < 8; ++v)
        dst[(mt * 16 + v + hi * 8) * 132 + n0 + nl] = c[v];
    }
    __syncthreads();
    float* t = src; src = dst; dst = t;  // src = Wh^(2^(s+1))
  }
  // src == Wh^64

  if (tid < KH) hv[tid] = 0.f;
  __syncthreads();
  for (int cc = 0; cc < KC; ++cc) {  // h = h @ Wh^64 + p_cc
    float acc = 0.f;
    if (tid < KH) {
      acc = Pws[(size_t)cc * KH + tid];
      for (int k = 0; k < KH; ++k) acc += hv[k] * src[k * 132 + tid];
    }
    __syncthreads();
    if (tid < KH) hv[tid] = acc;
    __syncthreads();
  }

  if (tid < KO) {
    float o = b2[tid];
    for (int n = 0; n < KH; ++n) o += hv[n] * W2[n * KO + tid];
    out[tid] = o;
  }
}

// ---------------------------------------------------------------------------
extern "C" void kernel_launch(void* const* d_in, const int* in_sizes, int n_in,
                              void* d_out, int out_size, void* d_ws, size_t ws_size,
                              hipStream_t stream) {
  const float* doc = (const float*)d_in[0];  // (T,1,1,E)
  const float* W1  = (const float*)d_in[1];  // (E+H, H)
  const float* b1  = (const float*)d_in[2];  // (H,)
  const float* W2  = (const float*)d_in[3];  // (H, O)
  const float* b2  = (const float*)d_in[4];  // (O,)
  float* out = (float*)d_out;                // (1, O)

  float* xproj = (float*)d_ws;               // T*H floats (2 MB)
  float* Pws   = xproj + (size_t)KT * KH;    // KC*H floats (32 KB)

  xproj_kernel<<<KT / 16, 256, 0, stream>>>(doc, W1, b1, xproj);
  chunk_scan_kernel<<<KC / 16, 256, 0, stream>>>(xproj, W1, Pws);
  const size_t ldsBytes = (size_t)(2 * 128 * 132 + 128) * sizeof(float);
  combine_kernel<<<1, 256, ldsBytes, stream>>>(W1, Pws, W2, b2, out);
}